// LinearAttentionAggregator_1949915152869
// MI455X (gfx1250) — compile-verified
//
#include <hip/hip_runtime.h>
#include <stdint.h>

// Problem dims (from reference)
#define T_DIM   4096
#define B_DIM   4
#define NH_DIM  16
#define HS_DIM  128
#define G_DIM   64
#define BH_DIM  (B_DIM*NH_DIM)        // 64 (b,h) pairs
#define COLS    (BH_DIM*HS_DIM)       // 8192 floats per timestep (contiguous)

// Chunked-scan config
#define CT      128                   // timesteps per chunk
#define LOG_CT  7
#define NCHUNK  (T_DIM/CT)            // 32
#define TS      16                    // rows per LDS tile (one WMMA M-tile)
#define NTILE   (CT/TS)               // 8
#define KROW    132                   // padded k-tile row (floats) -> 4-stride banks

typedef __attribute__((ext_vector_type(2))) float v2f;
typedef __attribute__((ext_vector_type(4))) float v4f;
typedef __attribute__((ext_vector_type(8))) float v8f;

// --- gfx1250 async DMA: global -> LDS, 16B per lane (one wave32 instr = 512B)
__device__ __forceinline__ void async_b128(uint32_t lds_byte_off, const void* src) {
  asm volatile("global_load_async_to_lds_b128 %0, %1, off"
               :: "v"(lds_byte_off), "v"(src) : "memory");
}
__device__ __forceinline__ uint32_t lds_off(const void* p) {
  // generic LDS address: low 32 bits are the wave-relative LDS byte offset
  return (uint32_t)(uintptr_t)p;
}

// Write in-chunk prefix snapshots for every boundary slot owned by (chunk, row gi).
// Slot j<G -> t0 = l[j]; slot j>=G -> t0 = r[j-G]+1. Owner chunk: t0>=T -> last.
__device__ __attribute__((noinline))
void snapshot_scan(int gi, int chunk, int bh, int lane,
                   const int* __restrict__ lidx, const int* __restrict__ ridx,
                   v4f accK, v4f accV,
                   float* __restrict__ bndK, float* __restrict__ bndV)
{
  const int t0base = chunk * CT;
  for (int j = 0; j < 2*G_DIM; ++j) {
    int t0 = (j < G_DIM) ? lidx[j] : (ridx[j - G_DIM] + 1);
    int c  = (t0 >= T_DIM) ? (NCHUNK - 1) : (t0 >> LOG_CT);
    if (c == chunk && (t0 - t0base) == gi) {
      size_t off = (size_t)j*COLS + (size_t)bh*HS_DIM + (size_t)lane*4;
      *(v4f*)(bndK + off) = accK;
      *(v4f*)(bndV + off) = accV;
    }
  }
}

// Pass 1: one wave32 per (chunk, bh). Streams CT rows of k and v through
// async-DMA double-buffered LDS tiles; WMMA computes the 16 dot products
// s = K_tile . qt per tile; VALU accumulates sum(k) and sum(s*v).
__global__ __launch_bounds__(32)
void pass1(const float* __restrict__ k, const float* __restrict__ v,
           const float* __restrict__ qt,
           const int* __restrict__ lidx, const int* __restrict__ ridx,
           float* __restrict__ chunkK, float* __restrict__ chunkV,
           float* __restrict__ bndK,   float* __restrict__ bndV)
{
  __shared__ float ktS[2][TS][KROW];    // padded: avoids 16-way bank conflicts on A-builds
  __shared__ float vtS[2][TS][HS_DIM];
  __shared__ float qtS[HS_DIM];

  const int chunk  = blockIdx.x;        // 0..NCHUNK-1
  const int bh     = blockIdx.y;        // 0..BH-1
  const int lane   = threadIdx.x;       // 0..31
  const int t0base = chunk * CT;

  // qt row for this (b,h): 512B, one async instruction
  async_b128(lds_off(&qtS[0]) + lane*16,
             (const char*)(qt + (size_t)bh*HS_DIM) + lane*16);

  auto issue_tile = [&](int tile, int buf) {
    const size_t trow0 = (size_t)(t0base + tile*TS);
    #pragma unroll
    for (int r = 0; r < TS; ++r) {
      const char* ksrc = (const char*)(k + (trow0 + r)*COLS + (size_t)bh*HS_DIM) + lane*16;
      const char* vsrc = (const char*)(v + (trow0 + r)*COLS + (size_t)bh*HS_DIM) + lane*16;
      async_b128(lds_off(&ktS[buf][r][0]) + lane*16, ksrc);
      async_b128(lds_off(&vtS[buf][r][0]) + lane*16, vsrc);
    }
  };

  issue_tile(0, 0);   // kick off tile 0 while we build the boundary mask

  // Uniform bitmask of boundary rows inside this chunk (rare hits)
  uint32_t bmask[CT/32] = {0u,0u,0u,0u};
  for (int j = 0; j < 2*G_DIM; ++j) {
    int t0 = (j < G_DIM) ? lidx[j] : (ridx[j - G_DIM] + 1);
    int c  = (t0 >= T_DIM) ? (NCHUNK - 1) : (t0 >> LOG_CT);
    if (c == chunk) {
      int i = t0 - t0base;
      if (i < CT) bmask[i >> 5] |= (1u << (i & 31));
    }
  }

  // A-matrix 16x4 f32 lane layout: lane m      -> A[m, d0+0], A[m, d0+1]
  //                                lane m+16   -> A[m, d0+2], A[m, d0+3]
  const uint32_t a_off = (uint32_t)((lane & 15)*(KROW*4) + ((lane >> 4) & 1)*8);
  const uint32_t b_off = (uint32_t)(((lane >> 4) & 1)*8);   // B: qt chunk broadcast over N

  v4f accK = {0.f,0.f,0.f,0.f};
  v4f accV = {0.f,0.f,0.f,0.f};

  for (int tile = 0; tile < NTILE; ++tile) {
    const int buf = tile & 1;
    if (tile + 1 < NTILE) {
      issue_tile(tile + 1, buf ^ 1);
      asm volatile("s_wait_asynccnt 32" ::: "memory");  // current tile (+qt) landed
    } else {
      asm volatile("s_wait_asynccnt 0" ::: "memory");
    }

    const char* ktb = (const char*)&ktS[buf][0][0];
    const char* vtb = (const char*)&vtS[buf][0][0];
    const char* qtb = (const char*)&qtS[0];

    // s_tile = K_tile(16x128) . qt via 32 accumulating f32 WMMAs
    v8f c8 = {0.f,0.f,0.f,0.f,0.f,0.f,0.f,0.f};
    #pragma unroll
    for (int d0 = 0; d0 < HS_DIM; d0 += 4) {
      v2f a = *(const v2f*)(ktb + a_off + (size_t)d0*4);
      v2f b = *(const v2f*)(qtb + b_off + (size_t)d0*4);
      c8 = __builtin_amdgcn_wmma_f32_16x16x4_f32(false, a, false, b,
                                                 (short)0, c8, false, false);
    }

    #pragma unroll
    for (int i = 0; i < TS; ++i) {
      const int gi = tile*TS + i;
      if ((bmask[gi >> 5] >> (gi & 31)) & 1u)   // wave-uniform, rare
        snapshot_scan(gi, chunk, bh, lane, lidx, ridx, accK, accV, bndK, bndV);
      // D layout: row m in element m (lanes 0-15), row m+8 in element m (lanes 16-31)
      int si = __builtin_amdgcn_readlane(__builtin_bit_cast(int, c8[i & 7]),
                                         (i < 8) ? 0 : 16);
      float s = __builtin_bit_cast(float, si);
      v4f kq = *(const v4f*)(ktb + (size_t)i*(KROW*4)   + lane*16);
      v4f vq = *(const v4f*)(vtb + (size_t)i*(HS_DIM*4) + lane*16);
      accK += kq;
      accV += s * vq;
    }
  }

  // boundaries sitting exactly at chunk end (only t0 == T for the last chunk)
  snapshot_scan(CT, chunk, bh, lane, lidx, ridx, accK, accV, bndK, bndV);

  const size_t co = (size_t)chunk*COLS + (size_t)bh*HS_DIM + (size_t)lane*4;
  *(v4f*)(chunkK + co) = accK;
  *(v4f*)(chunkV + co) = accV;
}

// Pass 2: prefix(t0) = sum_{c < chunk(t0)} S[c] + snapshot(t0);
// vhat = prefix(r+1)-prefix(l); khat = (prefixK(r+1)-prefixK(l)) / count.
__global__ __launch_bounds__(256)
void pass2(const int* __restrict__ lidx, const int* __restrict__ ridx,
           const float* __restrict__ chunkK, const float* __restrict__ chunkV,
           const float* __restrict__ bndK,   const float* __restrict__ bndV,
           float* __restrict__ out)
{
  const int tid = blockIdx.x*blockDim.x + threadIdx.x;  // G * COLS/4 threads
  const int g   = tid >> 11;                            // COLS/4 = 2048
  const int col = (tid & 2047) * 4;

  const int l  = lidx[g];
  const int r1 = ridx[g] + 1;
  const int cl = (l  >= T_DIM) ? (NCHUNK - 1) : (l  >> LOG_CT);
  const int cr = (r1 >= T_DIM) ? (NCHUNK - 1) : (r1 >> LOG_CT);

  const size_t bo_l = (size_t)g*COLS + col;
  const size_t bo_r = (size_t)(G_DIM + g)*COLS + col;
  v4f aK = *(const v4f*)(bndK + bo_r) - *(const v4f*)(bndK + bo_l);
  v4f aV = *(const v4f*)(bndV + bo_r) - *(const v4f*)(bndV + bo_l);
  for (int c = cl; c < cr; ++c) {
    const size_t o = (size_t)c*COLS + col;
    aK += *(const v4f*)(chunkK + o);
    aV += *(const v4f*)(chunkV + o);
  }
  const float inv = 1.0f / (float)(r1 - l);               // count = r-l+1
  *(v4f*)(out + (size_t)g*COLS + col)                       = aK * inv;  // khat
  *(v4f*)(out + (size_t)G_DIM*COLS + (size_t)g*COLS + col)  = aV;        // vhat
}

extern "C" void kernel_launch(void* const* d_in, const int* in_sizes, int n_in,
                              void* d_out, int out_size, void* d_ws, size_t ws_size,
                              hipStream_t stream)
{
  const float* k    = (const float*)d_in[0];
  const float* v    = (const float*)d_in[1];
  const float* qt   = (const float*)d_in[2];
  const int*   lidx = (const int*)d_in[3];
  const int*   ridx = (const int*)d_in[4];
  float* out = (float*)d_out;

  // workspace layout (floats): 10 MB total
  float* ws     = (float*)d_ws;
  float* chunkK = ws;                                   // NCHUNK*COLS
  float* chunkV = chunkK + (size_t)NCHUNK*COLS;         // NCHUNK*COLS
  float* bndK   = chunkV + (size_t)NCHUNK*COLS;         // 2G*COLS
  float* bndV   = bndK   + (size_t)2*G_DIM*COLS;        // 2G*COLS

  dim3 g1(NCHUNK, BH_DIM), b1(32);
  hipLaunchKernelGGL(pass1, g1, b1, 0, stream,
                     k, v, qt, lidx, ridx, chunkK, chunkV, bndK, bndV);

  dim3 g2((G_DIM*COLS/4)/256), b2(256);
  hipLaunchKernelGGL(pass2, g2, b2, 0, stream,
                     lidx, ridx, chunkK, chunkV, bndK, bndV, out);
}